// GNNModel_17600775979858
// MI455X (gfx1250) — compile-verified
//
#include <hip/hip_runtime.h>
#include <hip/hip_bf16.h>

typedef __attribute__((ext_vector_type(2))) float v2f;
typedef __attribute__((ext_vector_type(8))) float v8f;

// ---------------------------------------------------------------------------
// K1: deg[i] = 1 (self-loop)
__global__ void k_init_deg(float* __restrict__ deg, int N) {
    int i = blockIdx.x * blockDim.x + threadIdx.x;
    if (i < N) deg[i] = 1.0f;
}

// K2: deg[dst[e]] += 1
__global__ void k_count_deg(const int* __restrict__ dst, float* __restrict__ deg, int E) {
    int e = blockIdx.x * blockDim.x + threadIdx.x;
    if (e < E) atomicAdd(&deg[dst[e]], 1.0f);
}

// K3: deg -> dinv = rsqrt(deg)
__global__ void k_dinv(float* __restrict__ deg, int N) {
    int i = blockIdx.x * blockDim.x + threadIdx.x;
    if (i < N) {
        float d = deg[i];
        deg[i] = (d > 0.0f) ? rsqrtf(d) : 0.0f;
    }
}

// ---------------------------------------------------------------------------
// K4: h1[16-node tile] = x_tile(16x4) @ W1(4x16) via one V_WMMA_F32_16X16X4_F32
__global__ void k_gemm1_wmma(const float* __restrict__ x, const float* __restrict__ W1,
                             float* __restrict__ h1, int ntiles) {
    int wave = (blockIdx.x * blockDim.x + threadIdx.x) >> 5;
    int lane = threadIdx.x & 31;
    if (wave >= ntiles) return;              // wave-uniform: EXEC all-ones below

    int half = lane >> 4;                    // 0: lanes 0-15, 1: lanes 16-31
    int l    = lane & 15;                    // M for A, N for B/C/D
    int k0   = half * 2;                     // A/B hold K={k0,k0+1} in {x,y}

    v2f a, b;
    long node = (long)wave * 16 + l;
    a.x = x[node * 4 + k0];
    a.y = x[node * 4 + k0 + 1];
    b.x = W1[k0 * 16 + l];
    b.y = W1[(k0 + 1) * 16 + l];

    v8f c = {};
    c = __builtin_amdgcn_wmma_f32_16x16x4_f32(false, a, false, b, (short)0, c, false, false);

#pragma unroll
    for (int r = 0; r < 8; ++r) {
        int m = half ? (r + 8) : r;
        h1[((long)wave * 16 + m) * 16 + l] = c[r];
    }
}

// K5: agg1[n,f] = b1[f] + h1[n,f] * dinv[n]^2   (bias + self-loop folded in)
__global__ void k_init_agg1(float* __restrict__ agg1, const float* __restrict__ h1,
                            const float* __restrict__ dinv, const float* __restrict__ b1,
                            int N) {
    long gid = (long)blockIdx.x * blockDim.x + threadIdx.x;
    if (gid < (long)N * 16) {
        int n = (int)(gid >> 4);
        int f = (int)(gid & 15);
        float di = dinv[n];
        agg1[gid] = b1[f] + h1[gid] * di * di;
    }
}

// K6: for each edge: agg1[dst,f] += h1[src,f] * dinv[src]*dinv[dst]   (16 lanes/edge)
__global__ void k_edge_agg1(const int* __restrict__ src, const int* __restrict__ dst,
                            const float* __restrict__ h1, const float* __restrict__ dinv,
                            float* __restrict__ agg1, int E) {
    long gid = (long)blockIdx.x * blockDim.x + threadIdx.x;
    if (gid < (long)E * 16) {
        int e = (int)(gid >> 4);
        int f = (int)(gid & 15);
        int s = src[e], d = dst[e];
        float norm = dinv[s] * dinv[d];
        atomicAdd(&agg1[(long)d * 16 + f], h1[(long)s * 16 + f] * norm);
    }
}

// ---------------------------------------------------------------------------
// K7: h2[tile] = relu(agg1_tile 16x16) @ W2pad(16x16), 4 chained K=4 WMMAs.
//     W2 is 16x2; columns 2..15 of B are zero. Only N<2 of D is stored.
__global__ void k_gemm2_wmma(const float* __restrict__ agg1, const float* __restrict__ W2,
                             float* __restrict__ h2, int ntiles) {
    int wave = (blockIdx.x * blockDim.x + threadIdx.x) >> 5;
    int lane = threadIdx.x & 31;
    if (wave >= ntiles) return;              // wave-uniform

    int half = lane >> 4;
    int l    = lane & 15;

    v8f c = {};
#pragma unroll
    for (int kk = 0; kk < 4; ++kk) {
        int k0 = kk * 4 + half * 2;
        v2f a, b;
        float a0 = agg1[((long)wave * 16 + l) * 16 + k0];
        float a1 = agg1[((long)wave * 16 + l) * 16 + k0 + 1];
        a.x = a0 > 0.0f ? a0 : 0.0f;         // fused ReLU
        a.y = a1 > 0.0f ? a1 : 0.0f;
        b.x = (l < 2) ? W2[k0 * 2 + l] : 0.0f;
        b.y = (l < 2) ? W2[(k0 + 1) * 2 + l] : 0.0f;
        c = __builtin_amdgcn_wmma_f32_16x16x4_f32(false, a, false, b, (short)0, c, false, false);
    }

    if (l < 2) {
#pragma unroll
        for (int r = 0; r < 8; ++r) {
            int m = half ? (r + 8) : r;
            h2[((long)wave * 16 + m) * 2 + l] = c[r];
        }
    }
}

// K8: agg2[n,cl] = b2[cl] + h2[n,cl]*dinv[n]^2
__global__ void k_init_agg2(float* __restrict__ agg2, const float* __restrict__ h2,
                            const float* __restrict__ dinv, const float* __restrict__ b2,
                            int N) {
    long gid = (long)blockIdx.x * blockDim.x + threadIdx.x;
    if (gid < (long)N * 2) {
        int n  = (int)(gid >> 1);
        int cl = (int)(gid & 1);
        float di = dinv[n];
        agg2[gid] = b2[cl] + h2[gid] * di * di;
    }
}

// K9: per edge, 2 scatter-adds of h2[src]*norm into agg2[dst]
__global__ void k_edge_agg2(const int* __restrict__ src, const int* __restrict__ dst,
                            const float* __restrict__ h2, const float* __restrict__ dinv,
                            float* __restrict__ agg2, int E) {
    int e = blockIdx.x * blockDim.x + threadIdx.x;
    if (e < E) {
        int s = src[e], d = dst[e];
        float norm = dinv[s] * dinv[d];
        atomicAdd(&agg2[(long)d * 2 + 0], h2[(long)s * 2 + 0] * norm);
        atomicAdd(&agg2[(long)d * 2 + 1], h2[(long)s * 2 + 1] * norm);
    }
}

// K10: row-wise log_softmax over 2 classes
__global__ void k_logsoftmax(const float* __restrict__ agg2, float* __restrict__ out, int N) {
    int n = blockIdx.x * blockDim.x + threadIdx.x;
    if (n < N) {
        float v0 = agg2[(long)n * 2 + 0];
        float v1 = agg2[(long)n * 2 + 1];
        float m  = v0 > v1 ? v0 : v1;
        float lse = m + logf(expf(v0 - m) + expf(v1 - m));
        out[(long)n * 2 + 0] = v0 - lse;
        out[(long)n * 2 + 1] = v1 - lse;
    }
}

// Tail (N % 16 != 0) scalar fallbacks — not hit for N=1M but kept for safety.
__global__ void k_gemm1_tail(const float* __restrict__ x, const float* __restrict__ W1,
                             float* __restrict__ h1, int start, int N) {
    int t = blockIdx.x * blockDim.x + threadIdx.x;
    int n = start + (t >> 4), f = t & 15;
    if (n < N) {
        float acc = 0.0f;
        for (int k = 0; k < 4; ++k) acc += x[(long)n * 4 + k] * W1[k * 16 + f];
        h1[(long)n * 16 + f] = acc;
    }
}
__global__ void k_gemm2_tail(const float* __restrict__ agg1, const float* __restrict__ W2,
                             float* __restrict__ h2, int start, int N) {
    int t = blockIdx.x * blockDim.x + threadIdx.x;
    int n = start + (t >> 1), cl = t & 1;
    if (n < N) {
        float acc = 0.0f;
        for (int k = 0; k < 16; ++k) {
            float v = agg1[(long)n * 16 + k];
            acc += (v > 0.0f ? v : 0.0f) * W2[k * 2 + cl];
        }
        h2[(long)n * 2 + cl] = acc;
    }
}

// ---------------------------------------------------------------------------
extern "C" void kernel_launch(void* const* d_in, const int* in_sizes, int n_in,
                              void* d_out, int out_size, void* d_ws, size_t ws_size,
                              hipStream_t stream) {
    const float* x  = (const float*)d_in[0];
    const int*   ei = (const int*)d_in[1];     // edge_index [2,E] row-major
    const float* W1 = (const float*)d_in[3];
    const float* b1 = (const float*)d_in[4];
    const float* W2 = (const float*)d_in[5];
    const float* b2 = (const float*)d_in[6];
    float* out = (float*)d_out;

    const int N = in_sizes[0] / 4;
    const int E = in_sizes[1] / 2;
    const int* src = ei;
    const int* dst = ei + E;

    // Workspace layout: dinv | h1 | agg1 ; h2/agg2 alias dead h1 region.
    char* ws = (char*)d_ws;
    size_t off = 0;
    float* dinv = (float*)(ws + off); off += ((size_t)N * 4 + 255) & ~(size_t)255;
    float* h1   = (float*)(ws + off); size_t h1_off = off; off += (size_t)N * 16 * 4;
    float* agg1 = (float*)(ws + off);
    float* h2   = (float*)(ws + h1_off);                       // N*2 floats
    float* agg2 = (float*)(ws + h1_off + (size_t)N * 2 * 4);   // N*2 floats

    const int B = 256;
    const int gridN   = (N + B - 1) / B;
    const int gridE   = (E + B - 1) / B;
    const long n16    = (long)N * 16;
    const long e16    = (long)E * 16;
    const int gridN16 = (int)((n16 + B - 1) / B);
    const int gridE16 = (int)((e16 + B - 1) / B);
    const int gridN2  = (int)(((long)N * 2 + B - 1) / B);
    const int ntiles  = N / 16;
    const int rem     = N - ntiles * 16;
    const int gridT   = (ntiles + 7) / 8;      // 8 waves (wave32) per 256-thread block

    k_init_deg <<<gridN, B, 0, stream>>>(dinv, N);
    k_count_deg<<<gridE, B, 0, stream>>>(dst, dinv, E);
    k_dinv     <<<gridN, B, 0, stream>>>(dinv, N);

    if (ntiles > 0)
        k_gemm1_wmma<<<gridT, B, 0, stream>>>(x, W1, h1, ntiles);
    if (rem > 0)
        k_gemm1_tail<<<1, B, 0, stream>>>(x, W1, h1, ntiles * 16, N);

    k_init_agg1<<<gridN16, B, 0, stream>>>(agg1, h1, dinv, b1, N);
    k_edge_agg1<<<gridE16, B, 0, stream>>>(src, dst, h1, dinv, agg1, E);

    if (ntiles > 0)
        k_gemm2_wmma<<<gridT, B, 0, stream>>>(agg1, W2, h2, ntiles);
    if (rem > 0)
        k_gemm2_tail<<<1, B, 0, stream>>>(agg1, W2, h2, ntiles * 16, N);

    k_init_agg2<<<gridN2, B, 0, stream>>>(agg2, h2, dinv, b2, N);
    k_edge_agg2<<<gridE, B, 0, stream>>>(src, dst, h2, dinv, agg2, E);

    k_logsoftmax<<<gridN, B, 0, stream>>>(agg2, out, N);
}